// PerformerEncoder_22273700397102
// MI455X (gfx1250) — compile-verified
//
#include <hip/hip_runtime.h>
#include <hip/hip_bf16.h>
#include <math.h>

// ---------------------------------------------------------------------------
// Performer encoder, MI455X (gfx1250, wave32).
// Only logits[:,0,:] feed the outputs, so full-sequence work is limited to
// hn/K/V, the performer key features (dash/kp/ctx/kpsum) and the row-0 local
// attention keys/values. All heavy GEMMs use V_WMMA_F32_16X16X4_F32 (fp32).
// ---------------------------------------------------------------------------

typedef float v2f __attribute__((ext_vector_type(2)));
typedef float v8f __attribute__((ext_vector_type(8)));

#define NB   8
#define NN   2048
#define DIMM 64
#define INNER 512
#define NF   266        // NB_FEATURES = int(64*ln(64))
#define NFP  272        // padded to 17 WMMA tiles
#define ROWS (NB*NN)    // 16384
#define GROWS (NB*4*NN) // 65536 global-head feature rows

#define DN_C    0.3535533905932738f   // 64^-0.25
#define RATIO_C 0.0613139427f         // 266^-0.5
#define LOGE4_32 0.2878231366242557f  // ln(10000)/32

// workspace layout (floats)
#define OFF_HN   ((size_t)0)
#define OFF_KK   (OFF_HN  + (size_t)ROWS*DIMM)      //  1,048,576
#define OFF_VV   (OFF_KK  + (size_t)ROWS*INNER)     // +8,388,608
#define OFF_DASH (OFF_VV  + (size_t)ROWS*INNER)     // +8,388,608
#define OFF_CTX  (OFF_DASH+ (size_t)GROWS*NFP)      // +17,825,792
#define OFF_KPS  (OFF_CTX + (size_t)32*NFP*DIMM)    // +557,056
#define OFF_MX   (OFF_KPS + (size_t)32*NFP)         // +8,704

__device__ __forceinline__ unsigned enc_f(float f) {
    unsigned b = __float_as_uint(f);
    return (b & 0x80000000u) ? ~b : (b | 0x80000000u);
}
__device__ __forceinline__ float dec_f(unsigned u) {
    return (u & 0x80000000u) ? __uint_as_float(u ^ 0x80000000u)
                             : __uint_as_float(~u);
}

__global__ void init_mx_kernel(unsigned* mx) { *mx = 0x007FFFFFu; /* enc(-inf) */ }

// ---------------------------------------------------------------------------
// K0: x = tok_emb[ids] + fixed_pos_emb(n,64);  hn = LN(x, ln1).  1 wave/row.
// ---------------------------------------------------------------------------
__global__ void embed_ln_kernel(const int* __restrict__ ids,
                                const float* __restrict__ tok,
                                const float* __restrict__ g,
                                const float* __restrict__ be,
                                float* __restrict__ hn) {
    const int wave = threadIdx.x >> 5, lane = threadIdx.x & 31;
    const int row = blockIdx.x * 8 + wave;           // 0..16383
    const int n = row & (NN - 1);
    const int id = ids[row];
    float e[2];
#pragma unroll
    for (int q = 0; q < 2; ++q) {
        int j = lane + 32 * q;
        float ang = (float)n * __expf(-(float)(j & 31) * LOGE4_32);
        float pe = (j < 32) ? sinf(ang) : cosf(ang);
        e[q] = tok[id * DIMM + j] + pe;
    }
    float s = e[0] + e[1];
    for (int o = 16; o; o >>= 1) s += __shfl_xor(s, o, 32);
    float mu = s * (1.f / 64.f);
    float d0 = e[0] - mu, d1 = e[1] - mu;
    float vs = d0 * d0 + d1 * d1;
    for (int o = 16; o; o >>= 1) vs += __shfl_xor(vs, o, 32);
    float rstd = rsqrtf(vs * (1.f / 64.f) + 1e-5f);
    hn[(size_t)row * DIMM + lane]      = d0 * rstd * g[lane]      + be[lane];
    hn[(size_t)row * DIMM + lane + 32] = d1 * rstd * g[lane + 32] + be[lane + 32];
}

// ---------------------------------------------------------------------------
// K1: K = hn@Wk, V = hn@Wv  (16384x64 @ 64x512), WMMA f32 16x16x4.
// grid (1024 mtiles, 2 {K,V}), 8 waves, each wave owns 4 n-tiles (64 cols).
// ---------------------------------------------------------------------------
__global__ void gemm_kv_kernel(const float* __restrict__ hn,
                               const float* __restrict__ Wk,
                               const float* __restrict__ Wv,
                               float* __restrict__ kk,
                               float* __restrict__ vv) {
    const int mtile = blockIdx.x;
    const float* W = (blockIdx.y == 0) ? Wk : Wv;
    float* out = (blockIdx.y == 0) ? kk : vv;
    const int wave = threadIdx.x >> 5, lane = threadIdx.x & 31;
    const int half = lane >> 4, l16 = lane & 15;
    const int m = mtile * 16 + l16;

    v8f c[4];
#pragma unroll
    for (int j = 0; j < 4; ++j)
#pragma unroll
        for (int i = 0; i < 8; ++i) c[j][i] = 0.f;

    for (int ks = 0; ks < 16; ++ks) {
        const int kb = ks * 4 + half * 2;
        v2f a = *(const v2f*)(hn + (size_t)m * DIMM + kb);
#pragma unroll
        for (int j = 0; j < 4; ++j) {
            int col = (wave * 4 + j) * 16 + l16;
            v2f bf;
            bf.x = W[(size_t)kb * INNER + col];
            bf.y = W[(size_t)(kb + 1) * INNER + col];
            c[j] = __builtin_amdgcn_wmma_f32_16x16x4_f32(
                false, a, false, bf, (short)0, c[j], false, false);
        }
    }
#pragma unroll
    for (int j = 0; j < 4; ++j) {
        int col = (wave * 4 + j) * 16 + l16;
#pragma unroll
        for (int i = 0; i < 8; ++i)
            out[(size_t)(mtile * 16 + i + 8 * half) * INNER + col] = c[j][i];
    }
}

// ---------------------------------------------------------------------------
// K2: dash = (rotary(Kg)*dn) @ proj^T, store (dash - diag), atomicMax(dash).
// Rows r = (b*4+h)*2048+n (65536). grid (4096 mtiles, 17 ftiles), 1 wave.
// Rotary fused into A-frag: each lane's K-pair (kb even) is one rotary pair.
// ---------------------------------------------------------------------------
__global__ void dash_kernel(const float* __restrict__ kk,
                            const float* __restrict__ proj,
                            float* __restrict__ dashd,
                            unsigned* __restrict__ mxu) {
    const int mtile = blockIdx.x, ftile = blockIdx.y;
    const int lane = threadIdx.x & 31;
    const int half = lane >> 4, l16 = lane & 15;
    const int r = mtile * 16 + l16;
    const int bh = r >> 11, n = r & (NN - 1);
    const int b = bh >> 2, h = bh & 3;
    const size_t krow = (size_t)(b * NN + n) * INNER + h * 64;
    const int f = ftile * 16 + l16;
    const int fc = (f < NF) ? f : (NF - 1);   // clamp: duplicates don't change max

    float ssq = 0.f;
    v8f c;
#pragma unroll
    for (int i = 0; i < 8; ++i) c[i] = 0.f;

    for (int ks = 0; ks < 16; ++ks) {
        const int kb = ks * 4 + half * 2;
        v2f t = *(const v2f*)(kk + krow + kb);
        float ang = (float)n * __expf(-(float)(kb >> 1) * LOGE4_32);
        float sn, cs;
        sincosf(ang, &sn, &cs);
        v2f a;
        a.x = DN_C * (t.x * cs - t.y * sn);
        a.y = DN_C * (t.y * cs + t.x * sn);
        ssq += a.x * a.x + a.y * a.y;
        v2f bf = *(const v2f*)(proj + (size_t)fc * DIMM + kb);
        c = __builtin_amdgcn_wmma_f32_16x16x4_f32(
            false, a, false, bf, (short)0, c, false, false);
    }
    // global max of raw dash (reference: jnp.max(dash) for keys)
    float mx = c[0];
#pragma unroll
    for (int i = 1; i < 8; ++i) mx = fmaxf(mx, c[i]);
    for (int o = 16; o; o >>= 1) mx = fmaxf(mx, __shfl_xor(mx, o, 32));
    if (lane == 0) atomicMax(mxu, enc_f(mx));

    // diag per row: 0.5 * sum(a^2) (rotation preserves norm; a already * dn)
    float st = ssq + __shfl_xor(ssq, 16, 32);
    __shared__ float sd[16];
    if (lane < 16) sd[l16] = 0.5f * st;
    __syncthreads();
#pragma unroll
    for (int i = 0; i < 8; ++i) {
        int row = mtile * 16 + i + 8 * half;
        dashd[(size_t)row * NFP + f] = c[i] - sd[i + 8 * half];
    }
}

// ---------------------------------------------------------------------------
// K3: per (b,h): ctx = kp^T @ (Vg*mask)  (266x64, K=2048) and kpsum.
// kp = ratio*(exp(dashd - MX) + eps). grid (32 bh, 17 ftiles), 4 waves = d.
// ---------------------------------------------------------------------------
__global__ void ctx_kernel(const float* __restrict__ dashd,
                           const float* __restrict__ vv,
                           const int* __restrict__ ids,
                           const unsigned* __restrict__ mxu,
                           float* __restrict__ ctx,
                           float* __restrict__ kps) {
    const int bh = blockIdx.x, ftile = blockIdx.y;
    const int b = bh >> 2, h = bh & 3;
    const int wave = threadIdx.x >> 5, lane = threadIdx.x & 31;
    const int half = lane >> 4, l16 = lane & 15;
    const float MX = dec_f(*mxu);
    const int f = ftile * 16 + l16;
    const int vcol = h * 64 + wave * 16 + l16;
    const size_t dbase = (size_t)bh * NN * NFP;

    v8f c;
#pragma unroll
    for (int i = 0; i < 8; ++i) c[i] = 0.f;
    float kpa = 0.f;

    for (int st = 0; st < NN / 4; ++st) {
        const int n0 = st * 4 + half * 2;
        float d0 = dashd[dbase + (size_t)n0 * NFP + f];
        float d1 = dashd[dbase + (size_t)(n0 + 1) * NFP + f];
        v2f a;
        a.x = RATIO_C * (expf(d0 - MX) + 1e-4f);
        a.y = RATIO_C * (expf(d1 - MX) + 1e-4f);
        kpa += a.x + a.y;
        const size_t vr = (size_t)(b * NN + n0) * INNER + vcol;
        v2f bf;
        bf.x = (ids[b * NN + n0] != 0)     ? vv[vr]         : 0.f;
        bf.y = (ids[b * NN + n0 + 1] != 0) ? vv[vr + INNER] : 0.f;
        c = __builtin_amdgcn_wmma_f32_16x16x4_f32(
            false, a, false, bf, (short)0, c, false, false);
    }
#pragma unroll
    for (int i = 0; i < 8; ++i)
        ctx[((size_t)bh * NFP + ftile * 16 + i + 8 * half) * DIMM + wave * 16 + l16] = c[i];
    float tot = kpa + __shfl_xor(kpa, 16, 32);
    if (wave == 0 && lane < 16) kps[bh * NFP + ftile * 16 + l16] = tot;
}

// ---------------------------------------------------------------------------
// K4: everything at sequence position 0, one block per batch (256 threads).
// ---------------------------------------------------------------------------
__device__ __forceinline__ float brsum(float v, float* s, int tid) {
    s[tid] = v; __syncthreads();
    for (int o = 128; o > 0; o >>= 1) { if (tid < o) s[tid] += s[tid + o]; __syncthreads(); }
    float r = s[0]; __syncthreads(); return r;
}
__device__ __forceinline__ float brmax(float v, float* s, int tid) {
    s[tid] = v; __syncthreads();
    for (int o = 128; o > 0; o >>= 1) { if (tid < o) s[tid] = fmaxf(s[tid], s[tid + o]); __syncthreads(); }
    float r = s[0]; __syncthreads(); return r;
}

__global__ void final_kernel(const int* __restrict__ ids,
                             const float* __restrict__ tok,
                             const float* __restrict__ ln1g, const float* __restrict__ ln1b,
                             const float* __restrict__ Wq,  const float* __restrict__ Wo,
                             const float* __restrict__ ln2g, const float* __restrict__ ln2b,
                             const float* __restrict__ W1,  const float* __restrict__ b1,
                             const float* __restrict__ W2,  const float* __restrict__ b2,
                             const float* __restrict__ lnfg, const float* __restrict__ lnfb,
                             const float* __restrict__ Wout, const float* __restrict__ bout,
                             const float* __restrict__ Wc1, const float* __restrict__ bc1,
                             const float* __restrict__ Wc2, const float* __restrict__ bc2,
                             const float* __restrict__ proj,
                             const float* __restrict__ kk, const float* __restrict__ vv,
                             const float* __restrict__ ctx, const float* __restrict__ kps,
                             float* __restrict__ out) {
    __shared__ float s_red[256], s_part[256];
    __shared__ float s_x0[64], s_h[64], s_y1[64], s_y2[64];
    __shared__ float s_q0[512], s_cat[512], s_feat[NFP], s_qp[NFP];
    __shared__ float s_p[NN];
    __shared__ float s_a1[256];
    const int tid = threadIdx.x, b = blockIdx.x;
    const int base = b * NN;

    // x0 = tok_emb[id] + pos(0) : sin(0)=0, cos(0)=1
    if (tid < 64) s_x0[tid] = tok[ids[base] * DIMM + tid] + ((tid < 32) ? 0.f : 1.f);
    __syncthreads();
    // hn0 = LN(x0, ln1)
    float v = (tid < 64) ? s_x0[tid] : 0.f;
    float mu = brsum(v, s_red, tid) * (1.f / 64.f);
    float dv = (tid < 64) ? (s_x0[tid] - mu) : 0.f;
    float var = brsum(dv * dv, s_red, tid) * (1.f / 64.f);
    float rstd = rsqrtf(var + 1e-5f);
    if (tid < 64) s_h[tid] = dv * rstd * ln1g[tid] + ln1b[tid];
    __syncthreads();
    // q0 (rotary at n=0 is identity)
    for (int c = tid; c < INNER; c += 256) {
        float acc = 0.f;
        for (int j = 0; j < 64; ++j) acc += s_h[j] * Wq[j * INNER + c];
        s_q0[c] = acc;
    }
    __syncthreads();

    // ---- global (performer) heads ----
    for (int h = 0; h < 4; ++h) {
        for (int f = tid; f < NF; f += 256) {
            float acc = 0.f;
            for (int j = 0; j < 64; ++j) acc += s_q0[h * 64 + j] * proj[f * DIMM + j];
            s_feat[f] = acc * DN_C;
        }
        __syncthreads();
        float dq = (tid < 64) ? s_q0[h * 64 + tid] * s_q0[h * 64 + tid] : 0.f;
        float diag = brsum(dq, s_red, tid) * 0.0625f;   // 0.5*dn^2
        float lm = -1e30f;
        for (int f = tid; f < NF; f += 256) lm = fmaxf(lm, s_feat[f]);
        float mxq = brmax(lm, s_red, tid);
        float dacc = 0.f;
        for (int f = tid; f < NF; f += 256) {
            float qp = RATIO_C * (expf(s_feat[f] - diag - mxq) + 1e-4f);
            s_qp[f] = qp;
            dacc += qp * kps[(b * 4 + h) * NFP + f];
        }
        float denom = brsum(dacc, s_red, tid);
        __syncthreads();
        int d = tid & 63, part = tid >> 6;
        float acc = 0.f;
        for (int f = part; f < NF; f += 4)
            acc += s_qp[f] * ctx[((size_t)(b * 4 + h) * NFP + f) * DIMM + d];
        s_part[tid] = acc; __syncthreads();
        if (tid < 64)
            s_cat[h * 64 + tid] = (s_part[tid] + s_part[tid + 64] + s_part[tid + 128] + s_part[tid + 192]) / denom;
        __syncthreads();
    }

    // ---- local (softmax) heads, row 0 only ----
    for (int h = 4; h < 8; ++h) {
        float lm = -1e30f;
        for (int m = tid; m < NN; m += 256) {
            float acc = 0.f;
            const float* kr = kk + (size_t)(base + m) * INNER + h * 64;
            for (int j = 0; j < 64; ++j) acc += s_q0[h * 64 + j] * kr[j];
            acc *= 0.125f;                         // DIM_HEAD^-0.5
            if (ids[base + m] == 0) acc = -1e9f;
            s_p[m] = acc;
            lm = fmaxf(lm, acc);
        }
        float mx = brmax(lm, s_red, tid);
        float ls = 0.f;
        for (int m = tid; m < NN; m += 256) { float e = expf(s_p[m] - mx); s_p[m] = e; ls += e; }
        float ssum = brsum(ls, s_red, tid);
        __syncthreads();
        int d = tid & 63, part = tid >> 6;
        float acc = 0.f;
        for (int m = part * 512; m < part * 512 + 512; ++m)
            acc += s_p[m] * vv[(size_t)(base + m) * INNER + h * 64 + d];
        s_part[tid] = acc; __syncthreads();
        if (tid < 64)
            s_cat[h * 64 + tid] = (s_part[tid] + s_part[tid + 64] + s_part[tid + 128] + s_part[tid + 192]) / ssum;
        __syncthreads();
    }

    // attn_out @ Wo, residual
    if (tid < 64) {
        float acc = 0.f;
        for (int c = 0; c < INNER; ++c) acc += s_cat[c] * Wo[c * DIMM + tid];
        s_y1[tid] = s_x0[tid] + acc;
    }
    __syncthreads();
    // FF: LN(y1, ln2) -> gelu(W1)+b1 -> W2+b2, residual off x
    v = (tid < 64) ? s_y1[tid] : 0.f;
    mu = brsum(v, s_red, tid) * (1.f / 64.f);
    dv = (tid < 64) ? (s_y1[tid] - mu) : 0.f;
    var = brsum(dv * dv, s_red, tid) * (1.f / 64.f);
    rstd = rsqrtf(var + 1e-5f);
    if (tid < 64) s_h[tid] = dv * rstd * ln2g[tid] + ln2b[tid];
    __syncthreads();
    {
        float acc = b1[tid];
        for (int j = 0; j < 64; ++j) acc += s_h[j] * W1[j * 256 + tid];
        s_a1[tid] = 0.5f * acc * (1.f + erff(acc * 0.70710678118654752f));  // exact gelu
    }
    __syncthreads();
    if (tid < 64) {
        float acc = b2[tid];
        for (int i = 0; i < 256; ++i) acc += s_a1[i] * W2[i * DIMM + tid];
        s_y2[tid] = s_x0[tid] + acc;
    }
    __syncthreads();
    // h = LN(0.5*(y1+y2), lnf)
    float hv = (tid < 64) ? 0.5f * (s_y1[tid] + s_y2[tid]) : 0.f;
    mu = brsum(hv, s_red, tid) * (1.f / 64.f);
    dv = (tid < 64) ? (hv - mu) : 0.f;
    var = brsum(dv * dv, s_red, tid) * (1.f / 64.f);
    rstd = rsqrtf(var + 1e-5f);
    if (tid < 64) s_h[tid] = dv * rstd * lnfg[tid] + lnfb[tid];
    __syncthreads();
    // logits -> rep (output), classifier head
    if (tid < 64) {
        float acc = bout[tid];
        for (int j = 0; j < 64; ++j) acc += s_h[j] * Wout[j * 64 + tid];
        s_cat[tid] = acc;
        out[16 + b * 64 + tid] = acc;          // rep
    }
    __syncthreads();
    if (tid < 32) {
        float acc = bc1[tid];
        for (int j = 0; j < 64; ++j) acc += s_cat[j] * Wc1[j * 32 + tid];
        s_a1[tid] = fmaxf(acc, 0.f);
    }
    __syncthreads();
    if (tid < 2) {
        float acc = bc2[tid];
        for (int i = 0; i < 32; ++i) acc += s_a1[i] * Wc2[i * 2 + tid];
        out[b * 2 + tid] = acc;                // out
    }
}

// ---------------------------------------------------------------------------
extern "C" void kernel_launch(void* const* d_in, const int* in_sizes, int n_in,
                              void* d_out, int out_size, void* d_ws, size_t ws_size,
                              hipStream_t stream) {
    (void)in_sizes; (void)n_in; (void)out_size; (void)ws_size;
    const int*   ids   = (const int*)  d_in[0];
    const float* tok   = (const float*)d_in[1];
    const float* ln1g  = (const float*)d_in[2];
    const float* ln1b  = (const float*)d_in[3];
    const float* Wq    = (const float*)d_in[4];
    const float* Wk    = (const float*)d_in[5];
    const float* Wv    = (const float*)d_in[6];
    const float* Wo    = (const float*)d_in[7];
    const float* ln2g  = (const float*)d_in[8];
    const float* ln2b  = (const float*)d_in[9];
    const float* W1    = (const float*)d_in[10];
    const float* b1    = (const float*)d_in[11];
    const float* W2    = (const float*)d_in[12];
    const float* b2    = (const float*)d_in[13];
    const float* lnfg  = (const float*)d_in[14];
    const float* lnfb  = (const float*)d_in[15];
    const float* Wout  = (const float*)d_in[16];
    const float* bout  = (const float*)d_in[17];
    const float* Wc1   = (const float*)d_in[18];
    const float* bc1   = (const float*)d_in[19];
    const float* Wc2   = (const float*)d_in[20];
    const float* bc2   = (const float*)d_in[21];
    const float* proj  = (const float*)d_in[22];

    float* ws = (float*)d_ws;
    float* hn    = ws + OFF_HN;
    float* kk    = ws + OFF_KK;
    float* vv    = ws + OFF_VV;
    float* dashd = ws + OFF_DASH;
    float* ctx   = ws + OFF_CTX;
    float* kps   = ws + OFF_KPS;
    unsigned* mxu = (unsigned*)(ws + OFF_MX);
    float* out = (float*)d_out;

    init_mx_kernel<<<1, 1, 0, stream>>>(mxu);
    embed_ln_kernel<<<ROWS / 8, 256, 0, stream>>>(ids, tok, ln1g, ln1b, hn);
    gemm_kv_kernel<<<dim3(ROWS / 16, 2), 256, 0, stream>>>(hn, Wk, Wv, kk, vv);
    dash_kernel<<<dim3(GROWS / 16, NFP / 16), 32, 0, stream>>>(kk, proj, dashd, mxu);
    ctx_kernel<<<dim3(32, NFP / 16), 128, 0, stream>>>(dashd, vv, ids, mxu, ctx, kps);
    final_kernel<<<NB, 256, 0, stream>>>(ids, tok, ln1g, ln1b, Wq, Wo, ln2g, ln2b,
                                         W1, b1, W2, b2, lnfg, lnfb, Wout, bout,
                                         Wc1, bc1, Wc2, bc2, proj,
                                         kk, vv, ctx, kps, out);
}